// GATLayer_13529146982752
// MI455X (gfx1250) — compile-verified
//
#include <hip/hip_runtime.h>

#define HIDDEN 64
#define HEADS 4
#define CH (HEADS * HIDDEN) /* 256 */

typedef float v2f __attribute__((ext_vector_type(2)));
typedef float v8f __attribute__((ext_vector_type(8)));

// Float atomic max via sign-split ordering trick (monotone in both paths).
__device__ __forceinline__ void atomicMaxF32(float* addr, float value) {
    if (value >= 0.0f) {
        atomicMax((int*)addr, __float_as_int(value));
    } else {
        atomicMin((unsigned int*)addr, __float_as_uint(value));
    }
}

__device__ __forceinline__ void edge_sd(const int* __restrict__ ei, int e, int E,
                                        int& s, int& d) {
    if (e < E) { s = ei[e]; d = ei[E + e]; }
    else       { s = e - E; d = e - E; }      // appended self-loop
}

// K0: out[n*64+c] = bias[c];  m = -inf; denom = 0
__global__ void gat_k0_init(float* __restrict__ out, const float* __restrict__ bias,
                            float* __restrict__ m, float* __restrict__ denom, int N) {
    int id = blockIdx.x * blockDim.x + threadIdx.x;
    if (id < N * HIDDEN) out[id] = bias[id & (HIDDEN - 1)];
    if (id < N * HEADS) { m[id] = -__builtin_inff(); denom[id] = 0.0f; }
}

// K1: h(N,256) = x(N,64) @ W^T(64,256) via V_WMMA_F32_16X16X4_F32.
// One wave per 16x16 output tile, 16 WMMAs along K.
__global__ void gat_k1_gemm(const float* __restrict__ x, const float* __restrict__ W,
                            float* __restrict__ h, int N, int ntiles) {
    int tile = blockIdx.x * (blockDim.x >> 5) + (threadIdx.x >> 5);
    if (tile >= ntiles) return;                 // wave-uniform: EXEC all-1s inside
    int ct = tile & 15;                         // 256/16 col tiles
    int rt = tile >> 4;
    int lane  = threadIdx.x & 31;
    int l16   = lane & 15;
    int hi    = lane >> 4;                      // 0: K pair {0,1}, 1: K pair {2,3}
    int row   = rt * 16 + l16;                  // A rows, lanes 0-15 == 16-31
    int rowc  = row < N ? row : (N - 1);
    int bcol  = ct * 16 + l16;                  // B cols (W row index)
    const float* xrow = x + (size_t)rowc * HIDDEN;
    const float* wrow = W + (size_t)bcol * HIDDEN;

    v8f c = {};
    #pragma unroll
    for (int k0 = 0; k0 < HIDDEN; k0 += 4) {
        int k = k0 + hi * 2;
        v2f a, b;
        a[0] = xrow[k];     a[1] = xrow[k + 1];   // A 16x4 f32 layout (ISA 7.12.2)
        b[0] = wrow[k];     b[1] = wrow[k + 1];   // B 4x16: row=K striped, mirrored
        c = __builtin_amdgcn_wmma_f32_16x16x4_f32(
                /*neg_a=*/false, a, /*neg_b=*/false, b,
                /*c_mod=*/(short)0, c, /*reuse_a=*/false, /*reuse_b=*/false);
    }
    #pragma unroll
    for (int v = 0; v < 8; ++v) {               // C/D: VGPR v -> M = v + 8*hi
        int r = rt * 16 + v + hi * 8;
        if (r < N) h[(size_t)r * CH + ct * 16 + l16] = c[v];
    }
}

// K2: a_src[n,h] = <h[n,h,:], att_src[h,:]> ; same for a_dst. Wave per (n,h).
__global__ void gat_k2_att(const float* __restrict__ h,
                           const float* __restrict__ att_src,
                           const float* __restrict__ att_dst,
                           float* __restrict__ a_src, float* __restrict__ a_dst, int N) {
    int wid  = (blockIdx.x * blockDim.x + threadIdx.x) >> 5;
    int lane = threadIdx.x & 31;
    if (wid >= N * HEADS) return;
    int n = wid >> 2, hd = wid & 3;
    const float* hp = h + (size_t)n * CH + hd * HIDDEN;
    const float* as = att_src + hd * HIDDEN;
    const float* ad = att_dst + hd * HIDDEN;
    float s = hp[lane] * as[lane] + hp[lane + 32] * as[lane + 32];
    float d = hp[lane] * ad[lane] + hp[lane + 32] * ad[lane + 32];
    #pragma unroll
    for (int off = 16; off > 0; off >>= 1) {
        s += __shfl_xor(s, off, 32);
        d += __shfl_xor(d, off, 32);
    }
    if (lane == 0) { a_src[wid] = s; a_dst[wid] = d; }
}

// K3: segment max of leaky-relu logits, thread per (edge, head).
__global__ void gat_k3_max(const int* __restrict__ ei,
                           const float* __restrict__ a_src, const float* __restrict__ a_dst,
                           float* __restrict__ m, int E, int Etot) {
    int id = blockIdx.x * blockDim.x + threadIdx.x;
    if (id >= Etot * HEADS) return;
    int e = id >> 2, hd = id & 3;
    int s, d; edge_sd(ei, e, E, s, d);
    float l = a_src[s * HEADS + hd] + a_dst[d * HEADS + hd];
    l = l > 0.0f ? l : 0.2f * l;
    atomicMaxF32(&m[d * HEADS + hd], l);
}

// K4: denom[d,h] += exp(logit - m[d,h])
__global__ void gat_k4_denom(const int* __restrict__ ei,
                             const float* __restrict__ a_src, const float* __restrict__ a_dst,
                             const float* __restrict__ m, float* __restrict__ denom,
                             int E, int Etot) {
    int id = blockIdx.x * blockDim.x + threadIdx.x;
    if (id >= Etot * HEADS) return;
    int e = id >> 2, hd = id & 3;
    int s, d; edge_sd(ei, e, E, s, d);
    float l = a_src[s * HEADS + hd] + a_dst[d * HEADS + hd];
    l = l > 0.0f ? l : 0.2f * l;
    float ex = __expf(l - m[d * HEADS + hd]);
    atomicAdd(&denom[d * HEADS + hd], ex);
}

// K5: out[d, c] += sum_h (alpha_eh / H) * h[s, h, c].  Wave per edge; head is
// wave-uniform per iteration; 32 coalesced channel atomics per iteration.
__global__ void gat_k5_scatter(const int* __restrict__ ei, const float* __restrict__ h,
                               const float* __restrict__ a_src, const float* __restrict__ a_dst,
                               const float* __restrict__ m, const float* __restrict__ denom,
                               float* __restrict__ out, int E, int Etot) {
    int wid  = (blockIdx.x * blockDim.x + threadIdx.x) >> 5;
    int lane = threadIdx.x & 31;
    if (wid >= Etot) return;
    int s, d; edge_sd(ei, wid, E, s, d);
    const float* hs = h + (size_t)s * CH;
    float* od = out + (size_t)d * HIDDEN;
    float coef = 0.0f;
    #pragma unroll
    for (int it = 0; it < 8; ++it) {
        int hd = it >> 1;                        // uniform across the wave
        if ((it & 1) == 0) {
            float l = a_src[s * HEADS + hd] + a_dst[d * HEADS + hd];
            l = l > 0.0f ? l : 0.2f * l;
            float ex = __expf(l - m[d * HEADS + hd]);
            coef = ex / (denom[d * HEADS + hd] + 1e-16f) * 0.25f; // fold head-mean
        }
        int c = ((it & 1) << 5) + lane;
        atomicAdd(&od[c], coef * hs[hd * HIDDEN + c]);
    }
}

extern "C" void kernel_launch(void* const* d_in, const int* in_sizes, int n_in,
                              void* d_out, int out_size, void* d_ws, size_t ws_size,
                              hipStream_t stream) {
    const float* x       = (const float*)d_in[0];
    const int*   ei      = (const int*)d_in[1];
    const float* W       = (const float*)d_in[2];
    const float* att_src = (const float*)d_in[3];
    const float* att_dst = (const float*)d_in[4];
    const float* bias    = (const float*)d_in[5];
    float* out = (float*)d_out;

    const int N = in_sizes[0] / HIDDEN;     // 50000
    const int E = in_sizes[1] / 2;          // 800000
    const int Etot = E + N;                 // + self-loops

    // Workspace layout (floats): h | a_src | a_dst | m | denom  (~54.4 MB)
    float* ws     = (float*)d_ws;
    float* h      = ws;
    float* a_src  = h + (size_t)N * CH;
    float* a_dst  = a_src + (size_t)N * HEADS;
    float* m      = a_dst + (size_t)N * HEADS;
    float* denom  = m + (size_t)N * HEADS;

    // K0: init out (bias), m (-inf), denom (0)
    gat_k0_init<<<(N * HIDDEN + 255) / 256, 256, 0, stream>>>(out, bias, m, denom, N);

    // K1: WMMA GEMM, 8 waves (tiles) per 256-thread block
    int ntiles = ((N + 15) / 16) * (CH / 16);
    gat_k1_gemm<<<(ntiles + 7) / 8, 256, 0, stream>>>(x, W, h, N, ntiles);

    // K2: attention dot products, wave per (node, head)
    gat_k2_att<<<(N * HEADS * 32 + 255) / 256, 256, 0, stream>>>(h, att_src, att_dst,
                                                                 a_src, a_dst, N);

    // K3/K4: segment max, then denom
    int eh = Etot * HEADS;
    gat_k3_max<<<(eh + 255) / 256, 256, 0, stream>>>(ei, a_src, a_dst, m, E, Etot);
    gat_k4_denom<<<(eh + 255) / 256, 256, 0, stream>>>(ei, a_src, a_dst, m, denom, E, Etot);

    // K5: weighted scatter into out, wave per edge
    gat_k5_scatter<<<(Etot * 32 + 255) / 256, 256, 0, stream>>>(ei, h, a_src, a_dst,
                                                                m, denom, out, E, Etot);
}